// LuChipmanModel_87222195847935
// MI455X (gfx1250) — compile-verified
//
#include <hip/hip_runtime.h>
#include <math.h>

#define HW   262144
#define HH   512
#define WWd  512
#define NB   2
#define NCI  48
#define NCO  22
#define DEGf 57.29577951308232f

typedef __attribute__((ext_vector_type(2))) float v2f;
typedef __attribute__((ext_vector_type(8))) float v8f;

// ---------- orderable float <-> uint keys for atomic min/max ----------
__device__ __forceinline__ unsigned f2k(float f){
  unsigned u = __float_as_uint(f);
  return (u & 0x80000000u) ? ~u : (u | 0x80000000u);
}
__device__ __forceinline__ float k2f(unsigned k){
  unsigned u = (k & 0x80000000u) ? (k & 0x7fffffffu) : ~k;
  return __uint_as_float(u);
}

// ---------- small dense linear algebra ----------
__device__ __forceinline__ void inv4(const float a[4][4], float b[4][4]){
  float s0 = a[0][0]*a[1][1] - a[1][0]*a[0][1];
  float s1 = a[0][0]*a[1][2] - a[1][0]*a[0][2];
  float s2 = a[0][0]*a[1][3] - a[1][0]*a[0][3];
  float s3 = a[0][1]*a[1][2] - a[1][1]*a[0][2];
  float s4 = a[0][1]*a[1][3] - a[1][1]*a[0][3];
  float s5 = a[0][2]*a[1][3] - a[1][2]*a[0][3];
  float c5 = a[2][2]*a[3][3] - a[3][2]*a[2][3];
  float c4 = a[2][1]*a[3][3] - a[3][1]*a[2][3];
  float c3 = a[2][1]*a[3][2] - a[3][1]*a[2][2];
  float c2 = a[2][0]*a[3][3] - a[3][0]*a[2][3];
  float c1 = a[2][0]*a[3][2] - a[3][0]*a[2][2];
  float c0 = a[2][0]*a[3][1] - a[3][0]*a[2][1];
  float det = s0*c5 - s1*c4 + s2*c3 + s3*c2 - s4*c1 + s5*c0;
  float id = 1.0f/det;
  b[0][0] = ( a[1][1]*c5 - a[1][2]*c4 + a[1][3]*c3)*id;
  b[0][1] = (-a[0][1]*c5 + a[0][2]*c4 - a[0][3]*c3)*id;
  b[0][2] = ( a[3][1]*s5 - a[3][2]*s4 + a[3][3]*s3)*id;
  b[0][3] = (-a[2][1]*s5 + a[2][2]*s4 - a[2][3]*s3)*id;
  b[1][0] = (-a[1][0]*c5 + a[1][2]*c2 - a[1][3]*c1)*id;
  b[1][1] = ( a[0][0]*c5 - a[0][2]*c2 + a[0][3]*c1)*id;
  b[1][2] = (-a[3][0]*s5 + a[3][2]*s2 - a[3][3]*s1)*id;
  b[1][3] = ( a[2][0]*s5 - a[2][2]*s2 + a[2][3]*s1)*id;
  b[2][0] = ( a[1][0]*c4 - a[1][1]*c2 + a[1][3]*c0)*id;
  b[2][1] = (-a[0][0]*c4 + a[0][1]*c2 - a[0][3]*c0)*id;
  b[2][2] = ( a[3][0]*s4 - a[3][1]*s2 + a[3][3]*s0)*id;
  b[2][3] = (-a[2][0]*s4 + a[2][1]*s2 - a[2][3]*s0)*id;
  b[3][0] = (-a[1][0]*c3 + a[1][1]*c1 - a[1][2]*c0)*id;
  b[3][1] = ( a[0][0]*c3 - a[0][1]*c1 + a[0][2]*c0)*id;
  b[3][2] = (-a[3][0]*s3 + a[3][1]*s1 - a[3][2]*s0)*id;
  b[3][3] = ( a[2][0]*s3 - a[2][1]*s1 + a[2][2]*s0)*id;
}
__device__ __forceinline__ void mul4(const float a[4][4], const float b[4][4], float c[4][4]){
#pragma unroll
  for (int i=0;i<4;++i)
#pragma unroll
    for (int j=0;j<4;++j){
      float s = 0.f;
#pragma unroll
      for (int k=0;k<4;++k) s += a[i][k]*b[k][j];
      c[i][j] = s;
    }
}
__device__ __forceinline__ float det3(const float m[3][3]){
  return m[0][0]*(m[1][1]*m[2][2]-m[1][2]*m[2][1])
       - m[0][1]*(m[1][0]*m[2][2]-m[1][2]*m[2][0])
       + m[0][2]*(m[1][0]*m[2][1]-m[1][1]*m[2][0]);
}
__device__ __forceinline__ void inv3(const float m[3][3], float o[3][3]){
  float c00 = m[1][1]*m[2][2]-m[1][2]*m[2][1];
  float c01 = m[1][2]*m[2][0]-m[1][0]*m[2][2];
  float c02 = m[1][0]*m[2][1]-m[1][1]*m[2][0];
  float det = m[0][0]*c00 + m[0][1]*c01 + m[0][2]*c02;
  float id = 1.0f/det;
  o[0][0]=c00*id; o[0][1]=(m[0][2]*m[2][1]-m[0][1]*m[2][2])*id; o[0][2]=(m[0][1]*m[1][2]-m[0][2]*m[1][1])*id;
  o[1][0]=c01*id; o[1][1]=(m[0][0]*m[2][2]-m[0][2]*m[2][0])*id; o[1][2]=(m[0][2]*m[1][0]-m[0][0]*m[1][2])*id;
  o[2][0]=c02*id; o[2][1]=(m[0][1]*m[2][0]-m[0][0]*m[2][1])*id; o[2][2]=(m[0][0]*m[1][1]-m[0][1]*m[1][0])*id;
}

// ---------- kernel 1: init min/max keys ----------
__global__ void mm_init(unsigned* keys){
  int t = threadIdx.x;
  if (t < 2*NB) keys[t] = (t & 1) ? 0u : 0xFFFFFFFFu;   // even=min, odd=max
}

// ---------- kernel 2: per-batch min/max of channel 0 ----------
__global__ __launch_bounds__(256)
void mm_minmax(const float* __restrict__ x, unsigned* __restrict__ keys){
  int b = blockIdx.y;
  const float* p = x + (size_t)b*NCI*HW;   // channel 0 plane
  unsigned lmin = 0xFFFFFFFFu, lmax = 0u;
  for (int i = blockIdx.x*blockDim.x + threadIdx.x; i < HW; i += gridDim.x*blockDim.x){
    unsigned k = f2k(p[i]);
    lmin = lmin < k ? lmin : k;
    lmax = lmax > k ? lmax : k;
  }
  __shared__ unsigned smin[256], smax[256];
  int t = threadIdx.x;
  smin[t] = lmin; smax[t] = lmax;
  __syncthreads();
  for (int s = 128; s > 0; s >>= 1){
    if (t < s){
      smin[t] = smin[t] < smin[t+s] ? smin[t] : smin[t+s];
      smax[t] = smax[t] > smax[t+s] ? smax[t] : smax[t+s];
    }
    __syncthreads();
  }
  if (t == 0){
    atomicMin(&keys[b*2+0], smin[0]);
    atomicMax(&keys[b*2+1], smax[0]);
  }
}

// ---------- kernel 3: per-pixel Mueller pipeline ----------
__global__ __launch_bounds__(256)
void mm_main(const float* __restrict__ x, float* __restrict__ out,
             const unsigned* __restrict__ keys,
             float* __restrict__ cosp, float* __restrict__ sinp)
{
  int idx = blockIdx.x*256 + threadIdx.x;
  if (idx >= NB*HW) return;
  int b   = idx >> 18;
  int pix = idx & (HW-1);
  const float* xb = x + (size_t)b*NCI*HW + (size_t)pix;

  float Fm[4][4], Am[4][4], Wm[4][4];
#pragma unroll
  for (int k=0;k<16;++k){
    Fm[k>>2][k&3] = xb[(size_t)k*HW];
    Am[k>>2][k&3] = xb[(size_t)(k+16)*HW];
    Wm[k>>2][k&3] = xb[(size_t)(k+32)*HW];
  }
  float inten = Fm[0][0];

  // M = inv(A) * F * inv(W), normalized by M[0][0]
  float iA[4][4], iW[4][4], T0[4][4], M[4][4];
  inv4(Am, iA);
  inv4(Wm, iW);
  mul4(iA, Fm, T0);
  mul4(T0, iW, M);
  float invn = 1.0f / M[0][0];
#pragma unroll
  for (int i=0;i<4;++i)
#pragma unroll
    for (int j=0;j<4;++j) M[i][j] *= invn;

  // ---- outputs: inorm (ch0) + mchan (ch1..16) ----
  float imin = k2f(keys[b*2+0]);
  float imax = k2f(keys[b*2+1]);
  float* ob = out + (size_t)b*NCO*HW + (size_t)pix;
  ob[0] = (inten - imin) / (imax - imin);
#pragma unroll
  for (int k=0;k<16;++k) ob[(size_t)(1+k)*HW] = M[k>>2][k&3];

  // ---- charpoly realizability mask via coherency matrix (Hermitian 4x4) ----
  float Hre[4][4], Him[4][4];
  {
    const float q = 0.25f;
    Hre[0][0] = q*(M[0][0]+M[0][1]+M[1][0]+M[1][1]); Him[0][0]=0.f;
    Hre[1][1] = q*(M[0][0]-M[0][1]+M[1][0]-M[1][1]); Him[1][1]=0.f;
    Hre[2][2] = q*(M[0][0]+M[0][1]-M[1][0]-M[1][1]); Him[2][2]=0.f;
    Hre[3][3] = q*(M[0][0]-M[0][1]-M[1][0]+M[1][1]); Him[3][3]=0.f;
    Hre[0][1] = q*(M[0][2]+M[1][2]); Him[0][1] = -q*(M[0][3]+M[1][3]);
    Hre[0][2] = q*(M[2][0]+M[2][1]); Him[0][2] = -q*(M[3][0]+M[3][1]);
    Hre[0][3] = q*(M[2][2]-M[3][3]); Him[0][3] = -q*(M[2][3]+M[3][2]);
    Hre[1][2] = q*(M[2][2]+M[3][3]); Him[1][2] =  q*(M[2][3]-M[3][2]);
    Hre[1][3] = q*(M[2][0]-M[2][1]); Him[1][3] = -q*(M[3][0]-M[3][1]);
    Hre[2][3] = q*(M[0][2]-M[1][2]); Him[2][3] = -q*(M[0][3]-M[1][3]);
#pragma unroll
    for (int i=0;i<4;++i)
#pragma unroll
      for (int j=0;j<i;++j){ Hre[i][j] = Hre[j][i]; Him[i][j] = -Him[j][i]; }
  }
  float tr1 = Hre[0][0]+Hre[1][1]+Hre[2][2]+Hre[3][3];
  float H2r[4][4], H2i[4][4];
#pragma unroll
  for (int i=0;i<4;++i)
#pragma unroll
    for (int j=0;j<4;++j){
      float sr = 0.f, si = 0.f;
#pragma unroll
      for (int k=0;k<4;++k){
        sr += Hre[i][k]*Hre[k][j] - Him[i][k]*Him[k][j];
        si += Hre[i][k]*Him[k][j] + Him[i][k]*Hre[k][j];
      }
      H2r[i][j]=sr; H2i[i][j]=si;
    }
  float tr2 = H2r[0][0]+H2r[1][1]+H2r[2][2]+H2r[3][3];
  float tr3 = 0.f, tr4 = 0.f;
#pragma unroll
  for (int i=0;i<4;++i)
#pragma unroll
    for (int k=0;k<4;++k){
      tr3 += H2r[i][k]*Hre[k][i] - H2i[i][k]*Him[k][i];
      tr4 += H2r[i][k]*H2r[k][i] - H2i[i][k]*H2i[k][i];
    }
  float c2 = 0.5f*(tr1*tr1 - tr2);
  float c3 = (tr1*tr1*tr1 - 3.f*tr1*tr2 + 2.f*tr3)*(1.f/6.f);
  float c4 = 0.25f*(c3*tr1 - c2*tr2 + tr1*tr3 - tr4);   // Newton's identity: det
  bool mask = (tr1 >= 0.f) && (c2 >= 0.f) && (c3 >= 0.f) && (c4 >= 0.f);

  // ---- Lu-Chipman on (masked) M ----
  const float SAFE[4][4] = {{1.f,0.02f,0.01f,0.005f},
                            {0.f,0.95f,0.05f,0.f},
                            {0.f,-0.05f,0.9f,0.04f},
                            {0.f,0.f,-0.04f,0.85f}};
  float Ms[4][4];
#pragma unroll
  for (int i=0;i<4;++i)
#pragma unroll
    for (int j=0;j<4;++j) Ms[i][j] = mask ? M[i][j] : SAFE[i][j];

  float D0=Ms[0][1], D1=Ms[0][2], D2=Ms[0][3];
  float dd = D0*D0 + D1*D1 + D2*D2;
  float d2 = fminf(fmaxf(dd, 0.f), 1.f-1e-6f);
  float d  = sqrtf(d2 + 1e-12f);
  float sq = sqrtf(1.f - d2);
  float dm = fmaxf(d, 1e-6f);
  float Dh0=D0/dm, Dh1=D1/dm, Dh2=D2/dm;
  float om = 1.f - sq;

  float MD[4][4];
  MD[0][0]=1.f; MD[0][1]=D0; MD[0][2]=D1; MD[0][3]=D2;
  MD[1][0]=D0;  MD[2][0]=D1; MD[3][0]=D2;
  MD[1][1]=sq+om*Dh0*Dh0; MD[1][2]=om*Dh0*Dh1;     MD[1][3]=om*Dh0*Dh2;
  MD[2][1]=om*Dh1*Dh0;     MD[2][2]=sq+om*Dh1*Dh1; MD[2][3]=om*Dh1*Dh2;
  MD[3][1]=om*Dh2*Dh0;     MD[3][2]=om*Dh2*Dh1;    MD[3][3]=sq+om*Dh2*Dh2;

  float iMD[4][4]; inv4(MD, iMD);
  float Mp[4][4];                       // Mp = Ms * inv(MD)^T
#pragma unroll
  for (int i=0;i<4;++i)
#pragma unroll
    for (int j=0;j<4;++j){
      float s = 0.f;
#pragma unroll
      for (int k=0;k<4;++k) s += Ms[i][k]*iMD[j][k];
      Mp[i][j] = s;
    }

  float mp[3][3];
#pragma unroll
  for (int i=0;i<3;++i)
#pragma unroll
    for (int j=0;j<3;++j) mp[i][j] = Mp[i+1][j+1];
  float mmx[3][3];                      // mm = mp * mp^T (symmetric)
#pragma unroll
  for (int i=0;i<3;++i)
#pragma unroll
    for (int j=0;j<3;++j){
      float s = 0.f;
#pragma unroll
      for (int k=0;k<3;++k) s += mp[i][k]*mp[j][k];
      mmx[i][j] = s;
    }

  // closed-form 3x3 symmetric eigenvalues
  float qv = (mmx[0][0]+mmx[1][1]+mmx[2][2])*(1.f/3.f);
  float p1 = mmx[0][1]*mmx[0][1] + mmx[0][2]*mmx[0][2] + mmx[1][2]*mmx[1][2];
  float a0 = mmx[0][0]-qv, a1 = mmx[1][1]-qv, a2 = mmx[2][2]-qv;
  float p2 = a0*a0 + a1*a1 + a2*a2 + 2.f*p1;
  float l0, l1, l2;                     // ascending
  if (p2 <= 1e-30f){ l0 = l1 = l2 = qv; }
  else {
    float pv = sqrtf(p2*(1.f/6.f));
    float ip = 1.f/pv;
    float b00=a0*ip, b11=a1*ip, b22=a2*ip;
    float b01=mmx[0][1]*ip, b02=mmx[0][2]*ip, b12=mmx[1][2]*ip;
    float detB = b00*(b11*b22-b12*b12) - b01*(b01*b22-b12*b02) + b02*(b01*b12-b11*b02);
    float r = fminf(fmaxf(0.5f*detB, -1.f), 1.f);
    float phi = acosf(r)*(1.f/3.f);
    l2 = qv + 2.f*pv*cosf(phi);
    l0 = qv + 2.f*pv*cosf(phi + 2.0943951023931953f);
    l1 = 3.f*qv - l2 - l0;
  }
  float s1v = sqrtf(fmaxf(l2, 1e-12f));
  float s2v = sqrtf(fmaxf(l1, 1e-12f));
  float s3v = sqrtf(fmaxf(l0, 1e-12f));
  float sgn = (det3(mp) < 0.f) ? -1.f : 1.f;

  float e1s = s1v*s2v + s2v*s3v + s3v*s1v + 1e-6f;
  float tS  = s1v + s2v + s3v;
  float pS  = s1v*s2v*s3v;
  float A3[3][3], B3[3][3];
#pragma unroll
  for (int i=0;i<3;++i)
#pragma unroll
    for (int j=0;j<3;++j){
      A3[i][j] = mmx[i][j] + ((i==j) ? e1s : 0.f);
      B3[i][j] = tS*mmx[i][j] + ((i==j) ? pS : 0.f);
    }
  float iA3[3][3]; inv3(A3, iA3);
  float mdel[3][3];
#pragma unroll
  for (int i=0;i<3;++i)
#pragma unroll
    for (int j=0;j<3;++j){
      float s = 0.f;
#pragma unroll
      for (int k=0;k<3;++k) s += iA3[i][k]*B3[k][j];
      mdel[i][j] = sgn*s;
    }

  float inv1d = 1.f/(1.f - d2);
  float Pd[3];
#pragma unroll
  for (int i=0;i<3;++i)
    Pd[i] = (Ms[i+1][0] - (Ms[i+1][1]*D0 + Ms[i+1][2]*D1 + Ms[i+1][3]*D2))*inv1d;

  float G[4][4];
  G[0][0]=1.f+1e-6f; G[0][1]=0.f; G[0][2]=0.f; G[0][3]=0.f;
#pragma unroll
  for (int i=0;i<3;++i){
    G[i+1][0] = Pd[i];
#pragma unroll
    for (int j=0;j<3;++j) G[i+1][j+1] = mdel[i][j] + ((i==j) ? 1e-6f : 0.f);
  }
  float iG[4][4]; inv4(G, iG);
  float MR[4][4]; mul4(iG, Mp, MR);

  // ---- polarimetric outputs ----
  float totD = sqrtf(dd + 1e-12f);
  float mR00=MR[1][1], mR01=MR[1][2], mR10=MR[2][1], mR11=MR[2][2];
  float tlin = sqrtf((mR00+mR11)*(mR00+mR11) + (mR10-mR01)*(mR10-mR01) + 1e-12f) - 1.f;
  tlin = fminf(fmaxf(tlin, -1.f+1e-6f), 1.f-1e-6f);
  float linR = acosf(tlin)*DEGf;
  float Yc = MR[3][1]-MR[1][3];
  float Xc = MR[2][3]-MR[3][2];
  float azi = 0.5f*atan2f(Yc, Xc)*DEGf;
  azi = fmodf(azi, 180.f); if (azi < 0.f) azi += 180.f;
  float totP = 1.f - fabsf(mdel[0][0]+mdel[1][1]+mdel[2][2])*(1.f/3.f);

  ob[(size_t)17*HW] = linR;
  ob[(size_t)18*HW] = totP;
  ob[(size_t)19*HW] = totD;
  ob[(size_t)20*HW] = azi;

  // cos/sin of theta = azi * 2pi/180 == atan2(Yc,Xc): no sincos needed
  float rh = sqrtf(Xc*Xc + Yc*Yc);
  float cth, sth;
  if (rh > 0.f){ cth = Xc/rh; sth = Yc/rh; } else { cth = 1.f; sth = 0.f; }
  cosp[(size_t)b*HW + pix] = cth;
  sinp[(size_t)b*HW + pix] = sth;
}

// ---------- kernel 4: 8x8 box filter of cos/sin -> circular std (WMMA) ----------
// One wave per 16x16 output tile. S = Kv(16x24) * X(24x24 halo) * Kh^T(24x16),
// banded Kv/Kh generated per-lane, products via V_WMMA_F32_16X16X4_F32.
__global__ __launch_bounds__(32)
void mm_box(const float* __restrict__ cosp, const float* __restrict__ sinp,
            float* __restrict__ out)
{
  __shared__ float Xc[24][25];
  __shared__ float Xs[24][25];
  __shared__ float Tc[16][25];
  __shared__ float Ts[16][25];

  const int lane = threadIdx.x;
  const int b  = blockIdx.z;
  const int y0 = blockIdx.y*16, x0 = blockIdx.x*16;
  const float* cp = cosp + (size_t)b*HW;
  const float* sp = sinp + (size_t)b*HW;

  // load 23x23 halo (rows/cols 23 zero-padded); window = [-3,+4]
#pragma unroll
  for (int it=0; it<18; ++it){
    int e = lane + it*32;
    int r = e/24, c = e - r*24;
    int gy = y0 - 3 + r, gx = x0 - 3 + c;
    bool ok = (r < 23) && (c < 23) && (gy >= 0) && (gy < HH) && (gx >= 0) && (gx < WWd);
    float cv = 0.f, sv = 0.f;
    if (ok){ int o = gy*WWd + gx; cv = cp[o]; sv = sp[o]; }
    Xc[r][c] = cv; Xs[r][c] = sv;
  }
  __syncthreads();

  const int m  = lane & 15;
  const int hf = lane >> 4;

#if __has_builtin(__builtin_amdgcn_wmma_f32_16x16x4_f32)
  // vertical: T[m][col] = sum_{c=m..m+7} X[c][col]
#pragma unroll
  for (int pl=0; pl<2; ++pl){
    float (*X)[25] = pl ? Xs : Xc;
    float (*T)[25] = pl ? Ts : Tc;
#pragma unroll
    for (int cb=0; cb<2; ++cb){
      v8f acc = {};
      int col = cb*16 + m;              // B-fragment n = lane&15
#pragma unroll
      for (int j=0; j<6; ++j){
        v2f av, bv;
#pragma unroll
        for (int v=0; v<2; ++v){
          int c = 4*j + 2*hf + v;       // K index for both A and B fragments
          av[v] = (c >= m && c <= m+7) ? 1.0f : 0.0f;   // Kv[m][c]
          bv[v] = (col < 24) ? X[c][col] : 0.0f;
        }
        acc = __builtin_amdgcn_wmma_f32_16x16x4_f32(false, av, false, bv,
                                                    (short)0, acc, false, false);
      }
      if (col < 24){
#pragma unroll
        for (int v=0; v<8; ++v) T[hf*8+v][col] = acc[v];
      }
    }
  }
  __syncthreads();

  // horizontal: S[m][n] = sum_{c=n..n+7} T[m][c]
  v8f S[2];
#pragma unroll
  for (int pl=0; pl<2; ++pl){
    float (*T)[25] = pl ? Ts : Tc;
    v8f acc = {};
#pragma unroll
    for (int j=0; j<6; ++j){
      v2f av, bv;
#pragma unroll
      for (int v=0; v<2; ++v){
        int c = 4*j + 2*hf + v;
        av[v] = T[m][c];                               // A row = lane&15
        bv[v] = (c >= m && c <= m+7) ? 1.0f : 0.0f;    // Kh[n][c], n = lane&15
      }
      acc = __builtin_amdgcn_wmma_f32_16x16x4_f32(false, av, false, bv,
                                                  (short)0, acc, false, false);
    }
    S[pl] = acc;
  }

  float* obp = out + ((size_t)b*NCO + 21)*HW;
#pragma unroll
  for (int v=0; v<8; ++v){
    int gy = y0 + hf*8 + v, gx = x0 + m;               // C/D layout
    int ylo = max(gy-3, 0), yhi = min(gy+4, HH-1);
    int xlo = max(gx-3, 0), xhi = min(gx+4, WWd-1);
    float cnt = (float)((yhi-ylo+1)*(xhi-xlo+1));
    float Rc = S[0][v], Rs = S[1][v];
    float R = sqrtf(Rc*Rc + Rs*Rs + 1e-12f)/cnt;
    R = fminf(fmaxf(R, 1e-7f), 1.f-1e-7f);
    obp[gy*WWd + gx] = 0.5f*sqrtf(-2.f*logf(R))*DEGf;
  }
#else
  // scalar separable fallback
  for (int e = lane; e < 16*24; e += 32){
    int r = e/24, c = e - (e/24)*24;
    float sc = 0.f, ss = 0.f;
#pragma unroll
    for (int k=0;k<8;++k){ sc += Xc[r+k][c]; ss += Xs[r+k][c]; }
    Tc[r][c] = sc; Ts[r][c] = ss;
  }
  __syncthreads();
  float* obp = out + ((size_t)b*NCO + 21)*HW;
  for (int e = lane; e < 256; e += 32){
    int r = e >> 4, n = e & 15;
    float sc = 0.f, ss = 0.f;
#pragma unroll
    for (int k=0;k<8;++k){ sc += Tc[r][n+k]; ss += Ts[r][n+k]; }
    int gy = y0 + r, gx = x0 + n;
    int ylo = max(gy-3, 0), yhi = min(gy+4, HH-1);
    int xlo = max(gx-3, 0), xhi = min(gx+4, WWd-1);
    float cnt = (float)((yhi-ylo+1)*(xhi-xlo+1));
    float R = sqrtf(sc*sc + ss*ss + 1e-12f)/cnt;
    R = fminf(fmaxf(R, 1e-7f), 1.f-1e-7f);
    obp[gy*WWd + gx] = 0.5f*sqrtf(-2.f*logf(R))*DEGf;
  }
#endif
  (void)m; (void)hf;
}

// ---------- host-side launcher ----------
extern "C" void kernel_launch(void* const* d_in, const int* in_sizes, int n_in,
                              void* d_out, int out_size, void* d_ws, size_t ws_size,
                              hipStream_t stream)
{
  (void)in_sizes; (void)n_in; (void)out_size; (void)ws_size;
  const float* x = (const float*)d_in[0];
  float* out = (float*)d_out;

  // ws layout: [0,16B) min/max keys (2 per batch), then cos plane, then sin plane
  unsigned* keys = (unsigned*)d_ws;
  float* cosp = (float*)((char*)d_ws + 256);
  float* sinp = cosp + (size_t)NB*HW;

  mm_init  <<<1, 4, 0, stream>>>(keys);
  mm_minmax<<<dim3(64, NB), 256, 0, stream>>>(x, keys);
  mm_main  <<<(NB*HW)/256, 256, 0, stream>>>(x, out, keys, cosp, sinp);
  mm_box   <<<dim3(WWd/16, HH/16, NB), 32, 0, stream>>>(cosp, sinp, out);
}